// MultiSeq2SeqPredict_28011776704976
// MI455X (gfx1250) — compile-verified
//
#include <hip/hip_runtime.h>
#include <math.h>

#define Hh   1024
#define H3   3072
#define Vv   32000
#define LEN  64
#define Ee   3

// ---- workspace layout (float offsets) ----
#define OFF_XEMB   0            // E*L*H      = 196608
#define OFF_GIALL  196608       // E*L*3H     = 589824
#define OFF_HOUTS  786432       // E*L*H      = 196608
#define OFF_HSTATE 983040       // E*H        = 3072
#define OFF_GHENC  986112       // E*3H       = 9216
#define OFF_ENCOUT 995328       // L*H        = 65536
#define OFF_DH     1060864      // H
#define OFF_DH2    1061888      // H
#define OFF_CAT    1062912      // 2H
#define OFF_XV     1064960      // H
#define OFF_DGI    1065984      // 3H
#define OFF_DGH    1069056      // 3H
#define OFF_LOGZ   1072128      // LEN
#define OFF_PART   1072192      // pm[128], ps[128], pi[128]
#define OFF_SCAL   1072576      // ints: tok, done, cnt_enc, cnt_gru, cnt_log

typedef float v2f __attribute__((ext_vector_type(2)));
typedef float v8f __attribute__((ext_vector_type(8)));

__device__ __forceinline__ float sigf(float x) { return 1.f / (1.f + expf(-x)); }

__device__ __forceinline__ float dotN(const float* __restrict__ w,
                                      const float* __restrict__ x, int n4) {
  const float4* a = (const float4*)w;
  const float4* b = (const float4*)x;
  float s = 0.f;
  for (int i = 0; i < n4; ++i) {
    float4 p = a[i], q = b[i];
    s = fmaf(p.x, q.x, s); s = fmaf(p.y, q.y, s);
    s = fmaf(p.z, q.z, s); s = fmaf(p.w, q.w, s);
  }
  return s;
}

// "last block" barrier helper: returns true for exactly one block, after all
// blocks' global stores are device-visible.  Counter is reset for next use.
__device__ __forceinline__ bool lastBlockArrive(int* cnt, int nblk, int* sLast) {
  __threadfence();
  __syncthreads();
  if (threadIdx.x == 0) {
    int old = atomicAdd(cnt, 1);
    *sLast = (old == nblk - 1);
    if (*sLast) atomicExch(cnt, 0);
  }
  __syncthreads();
  if (!*sLast) return false;
  __threadfence();   // acquire: make other blocks' stores visible
  __syncthreads();
  return true;
}

// ---------------- init ----------------
__global__ void k_init(float* hstate, float* dh, int* st) {
  int i = blockIdx.x * blockDim.x + threadIdx.x;
  if (i < Ee * Hh) hstate[i] = 0.f;
  if (i < Hh) dh[i] = 0.f;
  if (i < 5) st[i] = 0;   // tok=SOS, done=0, counters=0
}

// ---------------- encoder: gather embeddings ----------------
__global__ void k_gather(const int* __restrict__ q, const int* __restrict__ c,
                         const float* __restrict__ emb, float* __restrict__ xemb) {
  int e = blockIdx.x / LEN, t = blockIdx.x % LEN;
  int id = (e == 0) ? q[t] : c[(e - 1) * LEN + t];
  const float* src = emb + ((size_t)e * Vv + id) * Hh;
  float* dst = xemb + ((size_t)e * LEN + t) * Hh;
  for (int i = threadIdx.x; i < Hh; i += blockDim.x) dst[i] = src[i];
}

// ---------------- encoder: gi = x @ W_ih^T + b_ih via fp32 WMMA ----------------
// M=64, N=3072, K=1024.  2x2 register tiling per wave: 4 WMMAs per 4 fragment
// loads (8 flop/byte).  Jobs: 3 * 2(ttp) * 96(nnp) = 576 waves = 72 blocks.
__global__ void k_enc_gemm_wmma(const float* __restrict__ xemb,
                                const float* __restrict__ Wih,
                                const float* __restrict__ bih,
                                float* __restrict__ gi_all) {
  int wave = blockIdx.x * (blockDim.x >> 5) + (threadIdx.x >> 5);
  int e   = wave / 192;
  int r2  = wave % 192;
  int ttp = r2 / 96;                 // time tile-pair (0..1) -> rows ttp*32
  int nnp = r2 % 96;                 // gate-row tile-pair   -> rows nnp*32
  int lane = threadIdx.x & 31;
  int hlf  = lane >> 4;
  int lid  = lane & 15;

  int t0 = ttp * 32, n0 = nnp * 32;
  const float* X0 = xemb + ((size_t)e * LEN + t0 + lid) * Hh + 2 * hlf;
  const float* X1 = X0 + (size_t)16 * Hh;
  const float* W0 = Wih + ((size_t)e * H3 + n0 + lid) * Hh + 2 * hlf;
  const float* W1 = W0 + (size_t)16 * Hh;

  v8f acc00 = {0.f,0.f,0.f,0.f,0.f,0.f,0.f,0.f};
  v8f acc01 = acc00, acc10 = acc00, acc11 = acc00;
  for (int k = 0; k < Hh; k += 4) {
    v2f a0 = *(const v2f*)(X0 + k);
    v2f a1 = *(const v2f*)(X1 + k);
    v2f b0 = *(const v2f*)(W0 + k);
    v2f b1 = *(const v2f*)(W1 + k);
    acc00 = __builtin_amdgcn_wmma_f32_16x16x4_f32(false, a0, false, b0, (short)0, acc00, false, false);
    acc01 = __builtin_amdgcn_wmma_f32_16x16x4_f32(false, a0, false, b1, (short)0, acc01, false, false);
    acc10 = __builtin_amdgcn_wmma_f32_16x16x4_f32(false, a1, false, b0, (short)0, acc10, false, false);
    acc11 = __builtin_amdgcn_wmma_f32_16x16x4_f32(false, a1, false, b1, (short)0, acc11, false, false);
  }
  float bias0 = bih[e * H3 + n0 + lid];
  float bias1 = bih[e * H3 + n0 + 16 + lid];
  float* base = gi_all + (size_t)e * LEN * H3;
#pragma unroll
  for (int r = 0; r < 8; ++r) {
    int ta = t0 + r + 8 * hlf;         // D: M = vgpr + 8*half, N = lane%16
    int tb = ta + 16;
    base[(size_t)ta * H3 + n0 + lid]      = acc00[r] + bias0;
    base[(size_t)ta * H3 + n0 + 16 + lid] = acc01[r] + bias1;
    base[(size_t)tb * H3 + n0 + lid]      = acc10[r] + bias0;
    base[(size_t)tb * H3 + n0 + 16 + lid] = acc11[r] + bias1;
  }
}

// ------- encoder recurrence: gh matvec + (last block) GRU gate update -------
__global__ void k_enc_rec(const float* __restrict__ Whh, const float* __restrict__ bhh,
                          const float* __restrict__ gi_all, float* __restrict__ gh,
                          float* __restrict__ hstate, float* __restrict__ houts,
                          int t, int* cnt) {
  __shared__ int sLast;
  int e = blockIdx.x / 12;
  int n = (blockIdx.x % 12) * 256 + threadIdx.x;
  gh[e * H3 + n] = dotN(Whh + ((size_t)e * H3 + n) * Hh, hstate + e * Hh, 256)
                   + bhh[e * H3 + n];
  if (!lastBlockArrive(cnt, Ee * 12, &sLast)) return;
  for (int idx = threadIdx.x; idx < Ee * Hh; idx += 256) {
    int e2 = idx >> 10, i = idx & (Hh - 1);
    const float* gi = gi_all + (size_t)e2 * LEN * H3 + (size_t)t * H3;
    const float* g2 = gh + e2 * H3;
    float h = hstate[idx];
    float r = sigf(gi[i] + g2[i]);
    float z = sigf(gi[Hh + i] + g2[Hh + i]);
    float nn = tanhf(gi[2 * Hh + i] + r * g2[2 * Hh + i]);
    float hn = (1.f - z) * nn + z * h;
    hstate[idx] = hn;
    houts[(size_t)e2 * LEN * Hh + (size_t)t * Hh + i] = hn;
  }
}

// ---------------- combine encoder outputs ----------------
__global__ void k_combine(const float* __restrict__ houts, float* __restrict__ encout) {
  size_t o = (size_t)blockIdx.x * Hh + threadIdx.x;
  encout[o] = houts[o] + 0.5f * (houts[(size_t)LEN * Hh + o] +
                                 houts[2 * (size_t)LEN * Hh + o]);
}

// ---------------- decoder: attention + context ----------------
__global__ void k_dec_attn(const float* __restrict__ dec_emb, const float* __restrict__ W_attn,
                           const float* __restrict__ b_attn, const float* __restrict__ enc_outs,
                           const float* __restrict__ hvecg, float* __restrict__ cat2,
                           const int* __restrict__ st, float* __restrict__ dout, int t) {
  __shared__ float ev[Hh];
  __shared__ float hv[Hh];
  __shared__ float sc[LEN];
  __shared__ float aw[LEN];
  int tid = threadIdx.x;
  int tok = st[0];
  ev[tid] = dec_emb[(size_t)tok * Hh + tid];
  hv[tid] = hvecg[tid];
  __syncthreads();
  int wv = tid >> 5, lane = tid & 31;
  for (int j = wv; j < LEN; j += 32) {
    const float* wr = W_attn + (size_t)j * 2 * Hh;
    float p = 0.f;
    for (int k = lane; k < 2 * Hh; k += 32) {
      float xk = (k < Hh) ? ev[k] : hv[k - Hh];
      p = fmaf(wr[k], xk, p);
    }
    for (int m = 16; m; m >>= 1) p += __shfl_xor(p, m, 32);
    if (lane == 0) sc[j] = p + b_attn[j];
  }
  __syncthreads();
  if (tid == 0) {
    float mx = sc[0];
    for (int j = 1; j < LEN; ++j) mx = fmaxf(mx, sc[j]);
    float s = 0.f;
    for (int j = 0; j < LEN; ++j) { float e2 = expf(sc[j] - mx); aw[j] = e2; s += e2; }
    float inv = 1.f / s;
    for (int j = 0; j < LEN; ++j) aw[j] *= inv;
  }
  __syncthreads();
  if (tid < LEN) dout[(size_t)LEN * Vv + (size_t)t * LEN + tid] = aw[tid];
  float acc = 0.f;
  for (int j = 0; j < LEN; ++j) acc = fmaf(aw[j], enc_outs[(size_t)j * Hh + tid], acc);
  cat2[tid] = ev[tid];
  cat2[Hh + tid] = acc;
}

// ---------------- decoder: x = relu(W_comb @ [e;applied] + b) ----------------
__global__ void k_dec_comb(const float* __restrict__ W_comb, const float* __restrict__ b_comb,
                           const float* __restrict__ cat2, float* __restrict__ xv) {
  int n = blockIdx.x * 256 + threadIdx.x;
  float v = dotN(W_comb + (size_t)n * 2 * Hh, cat2, 512) + b_comb[n];
  xv[n] = fmaxf(v, 0.f);
}

// ------ decoder: GRU matvecs + (last block) gates + freeze-on-done ------
__global__ void k_dec_gru(const float* __restrict__ Wih, const float* __restrict__ bih,
                          const float* __restrict__ Whh, const float* __restrict__ bhh,
                          const float* __restrict__ xv, float* __restrict__ hst,
                          float* __restrict__ gi, float* __restrict__ gh,
                          float* __restrict__ h2, const int* __restrict__ st, int* cnt) {
  __shared__ int sLast;
  int idx = blockIdx.x * 256 + threadIdx.x;   // 0..6143
  if (idx < H3) {
    gi[idx] = dotN(Wih + (size_t)idx * Hh, xv, 256) + bih[idx];
  } else {
    int n = idx - H3;
    gh[n] = dotN(Whh + (size_t)n * Hh, hst, 256) + bhh[n];
  }
  if (!lastBlockArrive(cnt, 24, &sLast)) return;
  int done = st[1];
  for (int i = threadIdx.x; i < Hh; i += 256) {
    float h = hst[i];
    float r = sigf(gi[i] + gh[i]);
    float z = sigf(gi[Hh + i] + gh[Hh + i]);
    float n = tanhf(gi[2 * Hh + i] + r * gh[2 * Hh + i]);
    float hn = (1.f - z) * n + z * h;
    h2[i] = hn;
    hst[i] = done ? h : hn;       // h_out = done ? h : h2 (old done)
  }
}

// -- decoder: logits (raw, to d_out) + partials + (last block) token update --
__global__ void k_logits(const float* __restrict__ W_out, const float* __restrict__ b_out,
                         const float* __restrict__ h2, float* __restrict__ pm,
                         float* __restrict__ ps, float* __restrict__ pi,
                         float* __restrict__ logZ, int* __restrict__ st,
                         float* __restrict__ dout, int t, int* cnt) {
  __shared__ float red[256];
  __shared__ int redi[256];
  __shared__ int sLast;
  int tid = threadIdx.x;
  int n = blockIdx.x * 256 + tid;
  float zz = dotN(W_out + (size_t)n * Hh, h2, 256) + b_out[n];
  dout[(size_t)t * Vv + n] = zz;          // raw logit; logZ subtracted at end
  red[tid] = zz; redi[tid] = n;
  __syncthreads();
  for (int s = 128; s; s >>= 1) {
    if (tid < s) {
      float a = red[tid], b = red[tid + s];
      int ia = redi[tid], ib = redi[tid + s];
      if (b > a || (b == a && ib < ia)) { red[tid] = b; redi[tid] = ib; }
    }
    __syncthreads();
  }
  float mb = red[0];
  int bidx = redi[0];
  __syncthreads();
  red[tid] = expf(zz - mb);
  __syncthreads();
  for (int s = 128; s; s >>= 1) {
    if (tid < s) red[tid] += red[tid + s];
    __syncthreads();
  }
  if (tid == 0) { pm[blockIdx.x] = mb; ps[blockIdx.x] = red[0]; pi[blockIdx.x] = (float)bidx; }
  if (!lastBlockArrive(cnt, 125, &sLast)) return;
  // reuse LDS for the 125 partials
  bool ok = tid < 125;
  red[tid]  = ok ? pm[tid] : -INFINITY;     // block maxes
  redi[tid] = ok ? (int)pi[tid] : 0;        // block argmax indices
  __shared__ float sum[128];
  if (tid < 128) sum[tid] = (tid < 125) ? ps[tid] : 0.f;
  __syncthreads();
  if (tid == 0) {
    float M = -INFINITY;
    for (int b = 0; b < 125; ++b) M = fmaxf(M, red[b]);
    float S = 0.f;
    for (int b = 0; b < 125; ++b) S += sum[b] * expf(red[b] - M);
    float bestv = -INFINITY; int arg = 0;
    for (int b = 0; b < 125; ++b)
      if (red[b] > bestv) { bestv = red[b]; arg = redi[b]; }
    logZ[t] = M + logf(S);
    int done = st[1], tok = st[0];
    int tok_out = done ? tok : arg;
    int new_done = done | (arg == 1);       // EOS == 1
    st[0] = tok_out; st[1] = new_done;
    dout[(size_t)LEN * Vv + LEN * LEN + t] = (float)tok_out;
  }
}

// ---------------- batched log-softmax finalization ----------------
__global__ void k_logp_final(float* __restrict__ dout, const float* __restrict__ logZ) {
  int t = blockIdx.y;
  int n = blockIdx.x * 256 + threadIdx.x;
  dout[(size_t)t * Vv + n] -= logZ[t];
}

extern "C" void kernel_launch(void* const* d_in, const int* in_sizes, int n_in,
                              void* d_out, int out_size, void* d_ws, size_t ws_size,
                              hipStream_t stream) {
  (void)in_sizes; (void)n_in; (void)out_size; (void)ws_size;
  const int*   qids    = (const int*)d_in[0];
  const int*   cids    = (const int*)d_in[1];
  const float* enc_emb = (const float*)d_in[2];
  const float* enc_Wih = (const float*)d_in[3];
  const float* enc_Whh = (const float*)d_in[4];
  const float* enc_bih = (const float*)d_in[5];
  const float* enc_bhh = (const float*)d_in[6];
  const float* dec_emb = (const float*)d_in[7];
  const float* W_attn  = (const float*)d_in[8];
  const float* b_attn  = (const float*)d_in[9];
  const float* W_comb  = (const float*)d_in[10];
  const float* b_comb  = (const float*)d_in[11];
  const float* dWih    = (const float*)d_in[12];
  const float* dWhh    = (const float*)d_in[13];
  const float* dbih    = (const float*)d_in[14];
  const float* dbhh    = (const float*)d_in[15];
  const float* W_out   = (const float*)d_in[16];
  const float* b_out   = (const float*)d_in[17];
  float* out = (float*)d_out;
  float* ws  = (float*)d_ws;

  float* xemb   = ws + OFF_XEMB;
  float* gi_all = ws + OFF_GIALL;
  float* houts  = ws + OFF_HOUTS;
  float* hstate = ws + OFF_HSTATE;
  float* ghenc  = ws + OFF_GHENC;
  float* encout = ws + OFF_ENCOUT;
  float* dh     = ws + OFF_DH;
  float* dh2    = ws + OFF_DH2;
  float* cat2   = ws + OFF_CAT;
  float* xv     = ws + OFF_XV;
  float* dgi    = ws + OFF_DGI;
  float* dgh    = ws + OFF_DGH;
  float* logZ   = ws + OFF_LOGZ;
  float* pm     = ws + OFF_PART;
  float* psum   = pm + 128;
  float* pidx   = pm + 256;
  int*   st     = (int*)(ws + OFF_SCAL);

  k_init<<<12, 256, 0, stream>>>(hstate, dh, st);
  k_gather<<<Ee * LEN, 256, 0, stream>>>(qids, cids, enc_emb, xemb);
  k_enc_gemm_wmma<<<72, 256, 0, stream>>>(xemb, enc_Wih, enc_bih, gi_all);
  for (int t = 0; t < LEN; ++t)
    k_enc_rec<<<Ee * 12, 256, 0, stream>>>(enc_Whh, enc_bhh, gi_all, ghenc,
                                           hstate, houts, t, st + 2);
  k_combine<<<LEN, 1024, 0, stream>>>(houts, encout);
  for (int t = 0; t < LEN; ++t) {
    k_dec_attn<<<1, 1024, 0, stream>>>(dec_emb, W_attn, b_attn, encout, dh, cat2, st, out, t);
    k_dec_comb<<<4, 256, 0, stream>>>(W_comb, b_comb, cat2, xv);
    k_dec_gru<<<24, 256, 0, stream>>>(dWih, dbih, dWhh, dbhh, xv, dh, dgi, dgh,
                                      dh2, st, st + 3);
    k_logits<<<125, 256, 0, stream>>>(W_out, b_out, dh2, pm, psum, pidx,
                                      logZ, st, out, t, st + 4);
  }
  k_logp_final<<<dim3(125, 64), 256, 0, stream>>>(out, logZ);
}